// NTXentLoss_36893769072681
// MI455X (gfx1250) — compile-verified
//
#include <hip/hip_runtime.h>
#include <math.h>

// ---------------------------------------------------------------------------
// NT-Xent loss: features [8192,1024] f32 -> scalar loss.
//   Z = row-normalize(features);  sim = Z Z^T * 10;  (T = 0.1)
//   loss = mean_i( logsumexp_j(sim_ij) - sim_i0 )
// Since sim_ii == 10 exactly (unit rows), rowmax == 10 -> fixed-shift LSE:
//   lse_i = 10 + log( sum_j exp(sim_ij - 10) )
// Fused GEMM (V_WMMA_F32_16X16X4_F32) + exp-accumulate; sim never stored.
// ---------------------------------------------------------------------------

typedef float v2f __attribute__((ext_vector_type(2)));
typedef float v4f __attribute__((ext_vector_type(4)));
typedef float v8f __attribute__((ext_vector_type(8)));

#define NROWS 8192
#define DIM   1024
#define TINV  10.0f   // 1 / temperature
#define RMAX  10.0f   // exact row max of sim (diagonal)

__global__ void zero_out_kernel(float* out) {
    if (threadIdx.x == 0 && blockIdx.x == 0) out[0] = 0.0f;
}

// One block per row: sum of squares -> scale -> write Z to workspace.
__global__ __launch_bounds__(256) void normalize_rows_kernel(
    const float* __restrict__ x, float* __restrict__ z) {
    const int row  = blockIdx.x;
    const int t    = threadIdx.x;
    const float* xr = x + (size_t)row * DIM;
    float*       zr = z + (size_t)row * DIM;

    v4f v = *(const v4f*)(xr + t * 4);           // 256 threads * 4 = 1024
    float ss = v.x * v.x + v.y * v.y + v.z * v.z + v.w * v.w;
    #pragma unroll
    for (int off = 16; off > 0; off >>= 1) ss += __shfl_xor(ss, off, 32);

    __shared__ float wsum[8];
    const int wave = t >> 5, lane = t & 31;
    if (lane == 0) wsum[wave] = ss;
    __syncthreads();
    float tot = 0.0f;
    #pragma unroll
    for (int i = 0; i < 8; ++i) tot += wsum[i];

    const float scale = 1.0f / fmaxf(sqrtf(tot), 1e-8f);
    v.x *= scale; v.y *= scale; v.z *= scale; v.w *= scale;
    *(v4f*)(zr + t * 4) = v;
}

// One block per 16-row M-tile. A-tile staged in LDS in WMMA fragment order:
//   Af[(k/4)*64 + half*32 + m*2 + c] = Z[m0+m][k + 2*half + c]
// so each lane's A fragment is one conflict-free ds_load_b64.
// f32 16x16x4 fragment layout (ISA 7.12.2): lane l -> M/N = l%16, K-pair
// selected by l/16; C/D: VGPR r holds (M=r, N=lane) / (M=r+8, N=lane-16).
__global__ __launch_bounds__(256) void sim_lse_kernel(
    const float* __restrict__ z, float* __restrict__ out) {
    __shared__ float Af[16 * DIM];   // 64 KB (LDS is 320 KB/WGP on CDNA5)
    __shared__ float Ssum[16];
    __shared__ float sim0[16];

    const int t  = threadIdx.x;
    const int m0 = blockIdx.x * 16;

    // Cooperative fill: coalesced global reads, fragment-order LDS writes.
    for (int i = t; i < 16 * (DIM / 4); i += 256) {
        const int m  = i >> 8;        // 0..15
        const int kq = i & 255;       // k/4
        v4f v = *(const v4f*)(z + (size_t)(m0 + m) * DIM + kq * 4);
        *(v2f*)&Af[kq * 64 +      m * 2] = (v2f){v.x, v.y};   // half 0: k, k+1
        *(v2f*)&Af[kq * 64 + 32 + m * 2] = (v2f){v.z, v.w};   // half 1: k+2, k+3
    }
    if (t < 16) Ssum[t] = 0.0f;
    __syncthreads();

    const int wave = t >> 5;
    const int lane = t & 31;
    const int half = lane >> 4;       // K-pair select
    const int lm   = lane & 15;       // M (for A) / N (for B) index

    float rowsum[8];
    #pragma unroll
    for (int r = 0; r < 8; ++r) rowsum[r] = 0.0f;

    // Each wave sweeps j-tiles wave, wave+8, ... (512 tiles / 8 waves = 64).
    for (int jt = wave; jt < NROWS / 16; jt += 8) {
        const int n0 = jt * 16;
        // B = Z^T: lane needs Z[n0+lm][k + 2*half .. +1]  (L2-resident b64)
        const float* Bb = z + (size_t)(n0 + lm) * DIM + 2 * half;

        v8f acc = {};
        #pragma unroll 4
        for (int k = 0; k < DIM; k += 4) {
            v2f a = *(const v2f*)&Af[(k >> 2) * 64 + half * 32 + lm * 2];
            v2f b = *(const v2f*)(Bb + k);
            acc = __builtin_amdgcn_wmma_f32_16x16x4_f32(
                false, a, false, b, (short)0, acc, false, false);
        }

        // Capture sim[:,0] (j-tile 0, N == 0 -> lanes 0 and 16 of wave 0).
        if (jt == 0 && lm == 0) {
            #pragma unroll
            for (int r = 0; r < 8; ++r) sim0[r + 8 * half] = acc[r] * TINV;
        }
        // exp-accumulate the whole 16x16 tile into per-row partials.
        #pragma unroll
        for (int r = 0; r < 8; ++r)
            rowsum[r] += __expf(acc[r] * TINV - RMAX);
    }

    // Reduce over the 16 N-lanes of each half-wave, then combine waves in LDS.
    #pragma unroll
    for (int r = 0; r < 8; ++r) {
        float s = rowsum[r];
        #pragma unroll
        for (int off = 1; off < 16; off <<= 1) s += __shfl_xor(s, off, 32);
        if (lm == 0) atomicAdd(&Ssum[r + 8 * half], s);
    }
    __syncthreads();

    if (t < 16) {
        const float contrib =
            (RMAX + logf(Ssum[t]) - sim0[t]) * (1.0f / (float)NROWS);
        atomicAdd(out, contrib);
    }
}

extern "C" void kernel_launch(void* const* d_in, const int* in_sizes, int n_in,
                              void* d_out, int out_size, void* d_ws, size_t ws_size,
                              hipStream_t stream) {
    (void)in_sizes; (void)n_in; (void)out_size; (void)ws_size;
    const float* features = (const float*)d_in[0];
    float* zbuf = (float*)d_ws;            // needs 8192*1024*4 = 32 MB scratch
    float* out  = (float*)d_out;

    zero_out_kernel<<<1, 32, 0, stream>>>(out);
    normalize_rows_kernel<<<NROWS, 256, 0, stream>>>(features, zbuf);
    sim_lse_kernel<<<NROWS / 16, 256, 0, stream>>>(zbuf, out);
}